// GNNBlock_59047210385689
// MI455X (gfx1250) — compile-verified
//
#include <hip/hip_runtime.h>
#include <math.h>

typedef float v2f __attribute__((ext_vector_type(2)));
typedef float v8f __attribute__((ext_vector_type(8)));

#define EPS_NORM 1e-12f
#define EPS_BN   1e-5f
#define LRELU    0.01f

__device__ __forceinline__ float fast_tanh(float x) {
#if __has_builtin(__builtin_amdgcn_tanhf)
  return __builtin_amdgcn_tanhf(x);   // v_tanh_f32 on gfx1250
#else
  return tanhf(x);
#endif
}

__device__ __forceinline__ v8f v8_splat(float x) {
  v8f r;
#pragma unroll
  for (int i = 0; i < 8; ++i) r[i] = x;
  return r;
}

// D = A(16x4 f32) * B(4x16 f32) + C(16x16 f32), one wave (32 lanes)
__device__ __forceinline__ v8f wmma4(v2f a, v2f b, v8f c) {
  return __builtin_amdgcn_wmma_f32_16x16x4_f32(
      /*neg_a=*/false, a, /*neg_b=*/false, b,
      /*c_mod=*/(short)0, c, /*reuse_a=*/false, /*reuse_b=*/false);
}

// ---------------------------------------------------------------------------
// Build the per-edge feature vector e = [ea0, ea1, |d|, d/|d| (8), pad0]
// for a group of 16 edges into LDS (lanes 0..15, one edge each).
// Edge features depend only on the ORIGINAL v (vres) -> identical both layers.
// ---------------------------------------------------------------------------
template <bool TAIL>
__device__ __forceinline__ void build_edge_lds(
    int lane, int base, int E,
    const float* __restrict__ vres, const int* __restrict__ eidx,
    const float* __restrict__ eattr,
    float (*e_lds)[12], int* s_lds, int* d_lds)
{
  if (lane < 16) {
    int e = base + lane;
    int s = 0, d = 0;
    float ea0 = 0.f, ea1 = 0.f;
    if (!TAIL || e < E) {
      s = eidx[e];
      d = eidx[E + e];
      ea0 = eattr[2 * e];
      ea1 = eattr[2 * e + 1];
    }
    if (s_lds) s_lds[lane] = s;
    if (d_lds) d_lds[lane] = d;
    const float4* ps = (const float4*)(vres + (size_t)s * 8);
    const float4* pd = (const float4*)(vres + (size_t)d * 8);
    float4 s0 = ps[0], s1 = ps[1];
    float4 dd0 = pd[0], dd1 = pd[1];
    float dx0 = dd0.x - s0.x, dx1 = dd0.y - s0.y, dx2 = dd0.z - s0.z, dx3 = dd0.w - s0.w;
    float dx4 = dd1.x - s1.x, dx5 = dd1.y - s1.y, dx6 = dd1.z - s1.z, dx7 = dd1.w - s1.w;
    float nsq = dx0*dx0 + dx1*dx1 + dx2*dx2 + dx3*dx3 +
                dx4*dx4 + dx5*dx5 + dx6*dx6 + dx7*dx7;
    float nn  = sqrtf(nsq);
    float inv = 1.0f / (nn + EPS_NORM);
    e_lds[lane][0] = ea0;
    e_lds[lane][1] = ea1;
    e_lds[lane][2] = nn;
    e_lds[lane][3]  = dx0 * inv;
    e_lds[lane][4]  = dx1 * inv;
    e_lds[lane][5]  = dx2 * inv;
    e_lds[lane][6]  = dx3 * inv;
    e_lds[lane][7]  = dx4 * inv;
    e_lds[lane][8]  = dx5 * inv;
    e_lds[lane][9]  = dx6 * inv;
    e_lds[lane][10] = dx7 * inv;
    e_lds[lane][11] = 0.f;   // K padding 11 -> 12
  }
}

// ---------------------------------------------------------------------------
// Utility kernels
// ---------------------------------------------------------------------------
__global__ void zero_kernel(float* __restrict__ p, int n) {
  int i = blockIdx.x * blockDim.x + threadIdx.x;
  int stride = gridDim.x * blockDim.x;
  for (; i < n; i += stride) p[i] = 0.f;
}

__global__ void count_kernel(const int* __restrict__ eidx,
                             float* __restrict__ cnt, int E) {
  int e = blockIdx.x * blockDim.x + threadIdx.x;
  if (e < E) unsafeAtomicAdd(&cnt[eidx[E + e]], 1.0f);
}

// ---------------------------------------------------------------------------
// Pass A group body: h = e @ W + b for 16 edges via 12 WMMAs, accumulate
// per-column sum(h), sum(h^2) into registers.
// ---------------------------------------------------------------------------
template <bool TAIL>
__device__ __forceinline__ void stats_group(
    int lane, int col, int kb, int hi8, int base, int E,
    const float* __restrict__ vres, const int* __restrict__ eidx,
    const float* __restrict__ eattr,
    const v2f (&bf)[3][4], const float (&bias)[4],
    float (*e_lds)[12], float (&s_acc)[4], float (&q_acc)[4])
{
  build_edge_lds<TAIL>(lane, base, E, vres, eidx, eattr, e_lds, nullptr, nullptr);
  __syncthreads();
  v2f a0, a1, a2;
  a0.x = e_lds[col][0 + kb]; a0.y = e_lds[col][1 + kb];
  a1.x = e_lds[col][4 + kb]; a1.y = e_lds[col][5 + kb];
  a2.x = e_lds[col][8 + kb]; a2.y = e_lds[col][9 + kb];
  __syncthreads();
#pragma unroll
  for (int nt = 0; nt < 4; ++nt) {
    v8f c = v8_splat(bias[nt]);
    c = wmma4(a0, bf[0][nt], c);
    c = wmma4(a1, bf[1][nt], c);
    c = wmma4(a2, bf[2][nt], c);
#pragma unroll
    for (int r = 0; r < 8; ++r) {
      if (!TAIL || (base + hi8 + r < E)) {
        float h = c[r];
        s_acc[nt] += h;
        q_acc[nt] = fmaf(h, h, q_acc[nt]);
      }
    }
  }
}

__global__ __launch_bounds__(32) void stats_kernel(
    const float* __restrict__ vres, const int* __restrict__ eidx,
    const float* __restrict__ eattr, const float* __restrict__ W,  // 11x64
    const float* __restrict__ b,                                   // 64
    float* __restrict__ accum,                                     // 128
    int E)
{
  const int lane = threadIdx.x;
  const int col  = lane & 15;
  const int kb   = (lane >> 4) * 2;   // K sub-offset per lane half
  const int hi8  = (lane >> 4) << 3;  // row offset of this lane's C rows

  // B fragments (weights), K padded with zeros beyond row 10
  v2f bf[3][4];
#pragma unroll
  for (int kt = 0; kt < 3; ++kt)
#pragma unroll
    for (int nt = 0; nt < 4; ++nt) {
      int k0 = kt * 4 + kb, k1 = k0 + 1;
      v2f t;
      t.x = (k0 < 11) ? W[k0 * 64 + nt * 16 + col] : 0.f;
      t.y = (k1 < 11) ? W[k1 * 64 + nt * 16 + col] : 0.f;
      bf[kt][nt] = t;
    }
  float bias[4];
#pragma unroll
  for (int nt = 0; nt < 4; ++nt) bias[nt] = b[nt * 16 + col];

  float s_acc[4] = {0.f, 0.f, 0.f, 0.f};
  float q_acc[4] = {0.f, 0.f, 0.f, 0.f};

  __shared__ __align__(16) float e_lds[16][12];

  const int nfull = E >> 4;   // groups with all 16 edges valid
  for (int g = blockIdx.x; g < nfull; g += gridDim.x)
    stats_group<false>(lane, col, kb, hi8, g << 4, E, vres, eidx, eattr,
                       bf, bias, e_lds, s_acc, q_acc);
  if ((E & 15) && blockIdx.x == 0)   // masked tail group (uniform branch)
    stats_group<true>(lane, col, kb, hi8, nfull << 4, E, vres, eidx, eattr,
                      bf, bias, e_lds, s_acc, q_acc);

  // lanes l and l+16 hold the same column -> pair-combine, then atomics
#pragma unroll
  for (int nt = 0; nt < 4; ++nt) {
    float s = s_acc[nt] + __shfl_xor(s_acc[nt], 16, 32);
    float q = q_acc[nt] + __shfl_xor(q_acc[nt], 16, 32);
    if (lane < 16) {
      unsafeAtomicAdd(&accum[nt * 16 + lane], s);
      unsafeAtomicAdd(&accum[64 + nt * 16 + lane], q);
    }
  }
}

// ---------------------------------------------------------------------------
// Fold BatchNorm into affine weights:  tanh(BN(e@W+b)) == tanh(e@W' + b')
// ---------------------------------------------------------------------------
__global__ __launch_bounds__(64) void affine_kernel(
    const float* __restrict__ accum, const float* __restrict__ W,
    const float* __restrict__ b, const float* __restrict__ g,
    const float* __restrict__ beta, float* __restrict__ Wp,  // 12x64
    float* __restrict__ bp,                                  // 64
    int E)
{
  int j = threadIdx.x;
  float invE = 1.0f / (float)E;
  float mu   = accum[j] * invE;
  float ms   = accum[64 + j] * invE;
  float var  = fmaxf(ms - mu * mu, 0.f);
  float s    = g[j] * rsqrtf(var + EPS_BN);
  float t    = beta[j] - mu * s;
#pragma unroll
  for (int k = 0; k < 11; ++k) Wp[k * 64 + j] = W[k * 64 + j] * s;
  Wp[11 * 64 + j] = 0.f;  // padded K row
  bp[j] = b[j] * s + t;
}

// ---------------------------------------------------------------------------
// Pass C group body: recompute h' via WMMA, w = tanh(h'),
// msg[e][o] = sum_i x[src[e]][i] * w[e][i*8+o], scatter-atomic into agg.
// WMMA C layout: lane l holds cols {c,c+16,c+32,c+48}, c=l&15 ->
// output o = c&7, i-set {b,b+2,b+4,b+6}, b=c>>3; lane l^8 has the other half.
// ---------------------------------------------------------------------------
template <bool TAIL>
__device__ __forceinline__ void msg_group(
    int lane, int col, int kb, int hi8, int bsel, int o, int base, int E,
    const float* __restrict__ x, const float* __restrict__ vres,
    const int* __restrict__ eidx, const float* __restrict__ eattr,
    const v2f (&bf)[3][4], const float (&bias)[4],
    float (*e_lds)[12], float (*x_lds)[8], int* s_lds, int* d_lds,
    float* __restrict__ agg)
{
  build_edge_lds<TAIL>(lane, base, E, vres, eidx, eattr, e_lds, s_lds, d_lds);
  __syncthreads();
  // stage x[src] for the 16 edges: each lane loads one float4
  {
    int m = lane >> 1, half = lane & 1;
    const float4* px = (const float4*)(x + (size_t)s_lds[m] * 8);
    float4 xv = px[half];
    ((float4*)(&x_lds[m][0]))[half] = xv;
  }
  v2f a0, a1, a2;
  a0.x = e_lds[col][0 + kb]; a0.y = e_lds[col][1 + kb];
  a1.x = e_lds[col][4 + kb]; a1.y = e_lds[col][5 + kb];
  a2.x = e_lds[col][8 + kb]; a2.y = e_lds[col][9 + kb];
  __syncthreads();

  float msgp[8];
#pragma unroll
  for (int r = 0; r < 8; ++r) msgp[r] = 0.f;

#pragma unroll
  for (int nt = 0; nt < 4; ++nt) {
    v8f c = v8_splat(bias[nt]);
    c = wmma4(a0, bf[0][nt], c);
    c = wmma4(a1, bf[1][nt], c);
    c = wmma4(a2, bf[2][nt], c);
    const int i = 2 * nt + bsel;
#pragma unroll
    for (int r = 0; r < 8; ++r) {
      float wv = fast_tanh(c[r]);
      msgp[r] = fmaf(wv, x_lds[hi8 + r][i], msgp[r]);
    }
  }
#pragma unroll
  for (int r = 0; r < 8; ++r) {
    float full = msgp[r] + __shfl_xor(msgp[r], 8, 32);
    int m = hi8 + r;
    if (((lane & 8) == 0) && (!TAIL || (base + m < E))) {
      unsafeAtomicAdd(&agg[(size_t)d_lds[m] * 8 + o], full);
    }
  }
  __syncthreads();  // LDS reuse fence before next group
}

__global__ __launch_bounds__(32) void msg_kernel(
    const float* __restrict__ x, const float* __restrict__ vres,
    const int* __restrict__ eidx, const float* __restrict__ eattr,
    const float* __restrict__ Wp, const float* __restrict__ bp,
    float* __restrict__ agg, int E)
{
  const int lane = threadIdx.x;
  const int col  = lane & 15;
  const int kb   = (lane >> 4) * 2;
  const int hi8  = (lane >> 4) << 3;
  const int bsel = col >> 3;
  const int o    = col & 7;

  v2f bf[3][4];
#pragma unroll
  for (int kt = 0; kt < 3; ++kt)
#pragma unroll
    for (int nt = 0; nt < 4; ++nt) {
      v2f t;
      t.x = Wp[(kt * 4 + kb) * 64 + nt * 16 + col];
      t.y = Wp[(kt * 4 + kb + 1) * 64 + nt * 16 + col];
      bf[kt][nt] = t;
    }
  float bias[4];
#pragma unroll
  for (int nt = 0; nt < 4; ++nt) bias[nt] = bp[nt * 16 + col];

  __shared__ __align__(16) float e_lds[16][12];
  __shared__ __align__(16) float x_lds[16][8];
  __shared__ int s_lds[16];
  __shared__ int d_lds[16];

  const int nfull = E >> 4;
  for (int g = blockIdx.x; g < nfull; g += gridDim.x)
    msg_group<false>(lane, col, kb, hi8, bsel, o, g << 4, E, x, vres, eidx,
                     eattr, bf, bias, e_lds, x_lds, s_lds, d_lds, agg);
  if ((E & 15) && blockIdx.x == 0)
    msg_group<true>(lane, col, kb, hi8, bsel, o, nfull << 4, E, x, vres, eidx,
                    eattr, bf, bias, e_lds, x_lds, s_lds, d_lds, agg);
}

// ---------------------------------------------------------------------------
// Node update: out = lrelu(agg/max(cnt,1) + x@root_w + root_b) [+ residual]
// ---------------------------------------------------------------------------
__global__ void node_kernel(
    const float* __restrict__ agg, const float* __restrict__ cnt,
    const float* __restrict__ x, const float* __restrict__ rw,
    const float* __restrict__ rb, const float* __restrict__ resid,
    float* __restrict__ out, int N)
{
  int n = blockIdx.x * blockDim.x + threadIdx.x;
  if (n >= N) return;
  float invc = 1.0f / fmaxf(cnt[n], 1.0f);
  const float4* px = (const float4*)(x + (size_t)n * 8);
  float4 x0 = px[0], x1v = px[1];
  float xv[8] = {x0.x, x0.y, x0.z, x0.w, x1v.x, x1v.y, x1v.z, x1v.w};
  const float* pa = agg + (size_t)n * 8;
  const float* pr = resid ? resid + (size_t)n * 8 : nullptr;
  float* po = out + (size_t)n * 8;
#pragma unroll
  for (int oo = 0; oo < 8; ++oo) {
    float acc = fmaf(pa[oo], invc, rb[oo]);
#pragma unroll
    for (int i = 0; i < 8; ++i) acc = fmaf(xv[i], rw[i * 8 + oo], acc);
    acc = (acc > 0.f) ? acc : LRELU * acc;
    if (pr) acc += pr[oo];
    po[oo] = acc;
  }
}

// ---------------------------------------------------------------------------
extern "C" void kernel_launch(void* const* d_in, const int* in_sizes, int n_in,
                              void* d_out, int out_size, void* d_ws, size_t ws_size,
                              hipStream_t stream)
{
  const float* v     = (const float*)d_in[0];
  const int*   eidx  = (const int*)d_in[1];
  const float* eattr = (const float*)d_in[2];
  const int N = in_sizes[0] / 8;
  const int E = in_sizes[2] / 2;

  const float* enw[2]  = {(const float*)d_in[3], (const float*)d_in[9]};
  const float* enb[2]  = {(const float*)d_in[4], (const float*)d_in[10]};
  const float* eng[2]  = {(const float*)d_in[5], (const float*)d_in[11]};
  const float* enbe[2] = {(const float*)d_in[6], (const float*)d_in[12]};
  const float* rw[2]   = {(const float*)d_in[7], (const float*)d_in[13]};
  const float* rb[2]   = {(const float*)d_in[8], (const float*)d_in[14]};

  float* ws    = (float*)d_ws;
  float* accum = ws;                    // 128 floats
  float* Wp    = ws + 128;              // 12*64
  float* bp    = ws + 896;              // 64
  float* cnt   = ws + 960;              // N
  float* agg   = cnt + N;               // 8N
  float* x1    = agg + (size_t)8 * N;   // 8N
  float* out   = (float*)d_out;

  const int statsGrid = 1536;
  const int msgGrid   = 4096;

  zero_kernel<<<256, 256, 0, stream>>>(cnt, N);
  count_kernel<<<(E + 255) / 256, 256, 0, stream>>>(eidx, cnt, E);

  const float* xin = v;
  for (int l = 0; l < 2; ++l) {
    zero_kernel<<<8, 64, 0, stream>>>(accum, 128);
    zero_kernel<<<1024, 256, 0, stream>>>(agg, 8 * N);
    stats_kernel<<<statsGrid, 32, 0, stream>>>(v, eidx, eattr, enw[l], enb[l],
                                               accum, E);
    affine_kernel<<<1, 64, 0, stream>>>(accum, enw[l], enb[l], eng[l], enbe[l],
                                        Wp, bp, E);
    msg_kernel<<<msgGrid, 32, 0, stream>>>(xin, v, eidx, eattr, Wp, bp, agg, E);
    if (l == 0) {
      node_kernel<<<(N + 255) / 256, 256, 0, stream>>>(agg, cnt, xin, rw[l],
                                                       rb[l], nullptr, x1, N);
      xin = x1;
    } else {
      node_kernel<<<(N + 255) / 256, 256, 0, stream>>>(agg, cnt, xin, rw[l],
                                                       rb[l], v, out, N);
    }
  }
}